// CellEncoder_27711128994328
// MI455X (gfx1250) — compile-verified
//
#include <hip/hip_runtime.h>
#include <hip/hip_bf16.h>

typedef __attribute__((ext_vector_type(2))) float v2f;
typedef __attribute__((ext_vector_type(4))) float v4f;
typedef __attribute__((ext_vector_type(8))) float v8f;

// ---------------------------------------------------------------------------
// Phase 0: pack W (K x N, row-major) into WMMA-B fragment order:
//   Wp[((k/4)*N + col)*4 + j] = W[k - k%4 + j][col],  j in 0..3
// After packing, a 4x16 B fragment is ONE coalesced global_load_b64 per lane:
// lanes 0-15 (kpair 0) read bytes [0,8) and lanes 16-31 (kpair 2) read bytes
// [8,16) of consecutive 16B col-chunks -> a fully dense 256B transaction.
// One-time cost: 512 KB read + 512 KB write.
// ---------------------------------------------------------------------------
__global__ void __launch_bounds__(256)
pack_w_kernel(const float* __restrict__ W, float* __restrict__ Wp, int K, int N) {
    const int i = blockIdx.x * blockDim.x + threadIdx.x;   // over (K/4)*N
    const int total = (K >> 2) * N;
    if (i >= total) return;
    const int col = i % N;
    const int kg  = i / N;
    v4f v;
#pragma unroll
    for (int j = 0; j < 4; ++j)
        v[j] = W[(size_t)(kg * 4 + j) * N + col];
    *(v4f*)(Wp + (size_t)i * 4) = v;
}

// ---------------------------------------------------------------------------
// Phase 1: per-cell segment mean via binary search over sorted segment_ids.
// One workgroup (64 threads = 2 waves) per cell; each thread owns 4 floats of
// the 256-dim embedding (float4 -> global_load_b128, fully coalesced).
// Each gathered row is read exactly once from HBM; reduction lives in VGPRs.
// ---------------------------------------------------------------------------
__device__ __forceinline__ int lower_bound_i32(const int* __restrict__ a, int n, int key) {
    int lo = 0, hi = n;
    while (lo < hi) {
        int mid = (lo + hi) >> 1;
        if (a[mid] < key) lo = mid + 1; else hi = mid;
    }
    return lo;
}

__global__ void __launch_bounds__(64)
seg_mean_kernel(const float* __restrict__ feats,      // (num_chunks, D)
                const int*   __restrict__ member_idx, // (num_members)
                const int*   __restrict__ seg_ids,    // (num_members) sorted
                float*       __restrict__ cell_emb,   // (num_cells, D) out
                int num_members, int D) {
    const int cell = blockIdx.x;
    const int tid  = threadIdx.x;          // 0..63, owns 4 consecutive floats

    const int lo = lower_bound_i32(seg_ids, num_members, cell);
    const int hi = lower_bound_i32(seg_ids, num_members, cell + 1);

    v4f acc = {0.f, 0.f, 0.f, 0.f};
    const int off = tid * 4;
    for (int m = lo; m < hi; ++m) {
        const int idx = member_idx[m];     // uniform across lanes -> broadcast
        acc += *(const v4f*)(feats + (size_t)idx * D + off);
    }
    const float inv = 1.0f / fmaxf((float)(hi - lo), 1.0f);  // empty cell -> 0
    *(v4f*)(cell_emb + (size_t)cell * D + off) = acc * inv;
}

// ---------------------------------------------------------------------------
// Phase 2: out = cell_emb (M x K) @ W (K x N) + b via V_WMMA_F32_16X16X4_F32.
// Wave macro-tile: 64 (M) x 64 (N) -> 4x4 grid of 16x16 accumulators.
// 8 waves per block cover N = 512; grid.x tiles M by 64.
// Per K-step: 4 A-frag b64 + 4 packed-B b64 loads feed 16 WMMAs.
// Uniform control flow in the K-loop (EXEC all-ones, as WMMA requires).
// ---------------------------------------------------------------------------
#define TM 4   // 16-row tiles per wave
#define TN 4   // 16-col tiles per wave

__global__ void __launch_bounds__(256)
gemm_bias_kernel(const float* __restrict__ A,     // (M, K) cell_emb
                 const float* __restrict__ Wp,    // packed (K/4, N, 4)
                 const float* __restrict__ bias,  // (N)
                 float*       __restrict__ out,   // (M, N)
                 int M, int K, int N) {
    const int lane  = threadIdx.x & 31;
    const int wave  = threadIdx.x >> 5;            // 0..7
    const int n0    = wave * (TN * 16);            // wave's 64-col slab
    const int m0    = blockIdx.x * (TM * 16);      // block's 64-row slab

    const int frow  = lane & 15;                   // A row / B col within tile
    const int kpair = (lane >> 4) * 2;             // K offset {0 or 2}

    v8f acc[TM][TN];
#pragma unroll
    for (int mt = 0; mt < TM; ++mt)
#pragma unroll
        for (int nt = 0; nt < TN; ++nt)
            acc[mt][nt] = (v8f){0.f,0.f,0.f,0.f,0.f,0.f,0.f,0.f};

    for (int k0 = 0; k0 < K; k0 += 4) {
        v2f afrag[TM];
#pragma unroll
        for (int mt = 0; mt < TM; ++mt) {
            int row = m0 + mt * 16 + frow;
            row = (row < M) ? row : 0;             // clamp; store is guarded
            afrag[mt] = *(const v2f*)(A + (size_t)row * K + k0 + kpair);
        }
        v2f bfrag[TN];
#pragma unroll
        for (int nt = 0; nt < TN; ++nt) {
            const int col = n0 + nt * 16 + frow;
            bfrag[nt] = *(const v2f*)(Wp + ((size_t)(k0 >> 2) * N + col) * 4 + kpair);
        }
#pragma unroll
        for (int mt = 0; mt < TM; ++mt)
#pragma unroll
            for (int nt = 0; nt < TN; ++nt)
                acc[mt][nt] = __builtin_amdgcn_wmma_f32_16x16x4_f32(
                    /*neg_a=*/false, afrag[mt],
                    /*neg_b=*/false, bfrag[nt],
                    /*c_mod=*/(short)0, acc[mt][nt],
                    /*reuse_a=*/false, /*reuse_b=*/false);
    }

    // Store: C/D layout -> VGPR r holds row (r + 8*lane[4]), col = lane[3:0]
    const int rhalf = (lane >> 4) * 8;
#pragma unroll
    for (int mt = 0; mt < TM; ++mt) {
#pragma unroll
        for (int nt = 0; nt < TN; ++nt) {
            const int col = n0 + nt * 16 + frow;
            const float bv = bias[col];
#pragma unroll
            for (int r = 0; r < 8; ++r) {
                const int row = m0 + mt * 16 + rhalf + r;
                if (row < M)
                    out[(size_t)row * N + col] = acc[mt][nt][r] + bv;
            }
        }
    }
}

// ---------------------------------------------------------------------------
// Host-side launcher. Inputs (reference order):
//   0: chunk_features (num_chunks*D f32)   1: member_idx (int32)
//   2: segment_ids (int32, sorted)         3: num_cells (scalar int)
//   4: W (D*N f32)                         5: b (N f32)
// Output: (num_cells, N) f32.
// Scratch layout in d_ws: [ Wp : K*N f32 | cell_emb : num_cells*D f32 ]
// ---------------------------------------------------------------------------
extern "C" void kernel_launch(void* const* d_in, const int* in_sizes, int n_in,
                              void* d_out, int out_size, void* d_ws, size_t ws_size,
                              hipStream_t stream) {
    const float* feats      = (const float*)d_in[0];
    const int*   member_idx = (const int*)  d_in[1];
    const int*   seg_ids    = (const int*)  d_in[2];
    const float* W          = (const float*)d_in[4];
    const float* bias       = (const float*)d_in[5];
    float*       out        = (float*)d_out;

    const int N           = in_sizes[5];          // output_dim (512)
    const int D           = in_sizes[4] / N;      // input_dim  (256)
    const int num_members = in_sizes[1];
    const int num_cells   = out_size / N;         // avoids device-scalar read

    float* Wp       = (float*)d_ws;               // K*N floats (512 KB)
    float* cell_emb = Wp + (size_t)D * N;         // num_cells*D floats

    // Phase 0: one-time W repack into B-fragment order
    const int pack_items = (D >> 2) * N;
    pack_w_kernel<<<(pack_items + 255) / 256, 256, 0, stream>>>(W, Wp, D, N);

    // Phase 1: segment mean (bandwidth bound, ~0.46 GB -> ~20 us at 23.3 TB/s)
    seg_mean_kernel<<<num_cells, 64, 0, stream>>>(
        feats, member_idx, seg_ids, cell_emb, num_members, D);

    // Phase 2: f32 WMMA GEMM + bias (13.1 GFLOP)
    const int m_blocks = (num_cells + (TM * 16) - 1) / (TM * 16);
    gemm_bias_kernel<<<m_blocks, 256, 0, stream>>>(
        cell_emb, Wp, bias, out, num_cells, D, N);
}